// Attention_84851373900515
// MI455X (gfx1250) — compile-verified
//
#include <hip/hip_runtime.h>
#include <hip/hip_bf16.h>
#include <math.h>

// Problem constants (from reference): B=4, T=12, R=2048, D=64, K=16
#define BB 4
#define TT 12
#define RR 2048
#define DD 64
#define KK 16
#define NROWS (BB * TT * RR)   // 98304 rows of 64 floats

typedef float v2f __attribute__((ext_vector_type(2)));
typedef float v8f __attribute__((ext_vector_type(8)));

__device__ __forceinline__ v8f wmma_f32_k4(v2f a, v2f b, v8f c) {
  // D = A(16x4,f32) * B(4x16,f32) + C(16x16,f32)
  return __builtin_amdgcn_wmma_f32_16x16x4_f32(
      /*neg_a=*/false, a, /*neg_b=*/false, b,
      /*c_mod=*/(short)0, c, /*reuse_a=*/false, /*reuse_b=*/false);
}

// ---------------------------------------------------------------------------
// Kernel 1: Q = X @ Wq ; Kp = X @ Wk   (GEMM 98304 x 64 x 64, fp32 WMMA)
// One wave per 16x16 output tile; 4 waves/block cover the 64 output columns
// of one 16-row M tile. A-fragment is loaded once and feeds BOTH accumulators.
// ---------------------------------------------------------------------------
__global__ __launch_bounds__(128)
void proj_qk_kernel(const float* __restrict__ X,
                    const float* __restrict__ Wq,
                    const float* __restrict__ Wk,
                    float* __restrict__ Q,
                    float* __restrict__ Kp) {
  const int lane  = threadIdx.x & 31;
  const int wave  = threadIdx.x >> 5;        // 0..3 -> N tile
  const int m0    = blockIdx.x << 4;         // 16-row tile
  const int n0    = wave << 4;
  const int l15   = lane & 15;
  const int khalf = (lane >> 4) << 1;        // 0 for lanes 0-15, 2 for 16-31
  const int mrow  = m0 + l15;
  const int ncol  = n0 + l15;

  v8f accq = {};
  v8f acck = {};

#pragma unroll
  for (int k0 = 0; k0 < DD; k0 += 4) {
    const int krow = k0 + khalf;
    // A fragment: lane holds X[mrow, krow] and X[mrow, krow+1] (contiguous)
    v2f a;
    const float* ap = X + mrow * DD + krow;
    a.x = ap[0];
    a.y = ap[1];
    // B fragments: lane holds W[krow, ncol] and W[krow+1, ncol]
    v2f bq, bk;
    bq.x = Wq[krow * DD + ncol];
    bq.y = Wq[(krow + 1) * DD + ncol];
    bk.x = Wk[krow * DD + ncol];
    bk.y = Wk[(krow + 1) * DD + ncol];
    accq = wmma_f32_k4(a, bq, accq);
    acck = wmma_f32_k4(a, bk, acck);
  }

  // C/D layout: VGPR i -> (M = i or 8+i depending on lane half, N = lane&15)
  const int mbase = m0 + ((lane >> 4) << 3);
#pragma unroll
  for (int i = 0; i < 8; ++i) {
    Q [(mbase + i) * DD + ncol] = accq[i];
    Kp[(mbase + i) * DD + ncol] = acck[i];
  }
}

// ---------------------------------------------------------------------------
// Kernel 2: attention core, one wave32 per row.
// Each lane owns 2 consecutive dims (float2) of the 64-wide row.
// scores_k = q . Kp[neigh_k]  via butterfly shfl_xor reduction (result
// broadcast to all lanes), softmax over 16 scores replicated per lane,
// ctx from register-resident neighbor fragments, out = q + ctx in-place.
// ---------------------------------------------------------------------------
__global__ __launch_bounds__(256)
void attn_kernel(float* __restrict__ qo,          // Q in, (q+ctx) out, in-place
                 const float* __restrict__ kbuf,  // Kp
                 const int* __restrict__ nidx) {  // (R, K) neighbor indices
  const int wavesPerBlock = blockDim.x >> 5;
  const int row  = blockIdx.x * wavesPerBlock + (threadIdx.x >> 5);
  const int lane = threadIdx.x & 31;
  if (row >= NROWS) return;

  const int r      = row & (RR - 1);   // R is a power of two
  const int btBase = row - r;          // (b*T + t) * R

  float* qp = qo + (size_t)row * DD + (lane << 1);
  float qx = qp[0];
  float qy = qp[1];

  float kvx[KK], kvy[KK], sc[KK];

#pragma unroll
  for (int k = 0; k < KK; ++k) {
    const int nrow = btBase + nidx[r * KK + k];
    const float* kp = kbuf + (size_t)nrow * DD + (lane << 1);
    const float kx = kp[0];
    const float ky = kp[1];
    kvx[k] = kx;
    kvy[k] = ky;
    float p = qx * kx + qy * ky;
    p += __shfl_xor(p, 16, 32);
    p += __shfl_xor(p,  8, 32);
    p += __shfl_xor(p,  4, 32);
    p += __shfl_xor(p,  2, 32);
    p += __shfl_xor(p,  1, 32);
    sc[k] = p;                         // full score broadcast to every lane
  }

  // softmax over the 16 scores (replicated per lane; numerically stable)
  float m = sc[0];
#pragma unroll
  for (int k = 1; k < KK; ++k) m = fmaxf(m, sc[k]);
  float s = 0.f;
#pragma unroll
  for (int k = 0; k < KK; ++k) {
    const float e = __expf(sc[k] - m);
    sc[k] = e;
    s += e;
  }
  const float inv = 1.f / s;

  float cx = 0.f, cy = 0.f;
#pragma unroll
  for (int k = 0; k < KK; ++k) {
    const float a = sc[k] * inv;
    cx = fmaf(a, kvx[k], cx);
    cy = fmaf(a, kvy[k], cy);
  }

  qp[0] = qx + cx;                     // out = q + ctx (in-place)
  qp[1] = qy + cy;
}

// ---------------------------------------------------------------------------
// Kernel 3: result = sigmoid(Out @ Wd)   (GEMM + epilogue, fp32 WMMA)
// ---------------------------------------------------------------------------
__global__ __launch_bounds__(128)
void out_proj_sigmoid_kernel(const float* __restrict__ X,
                             const float* __restrict__ W,
                             float* __restrict__ Yout) {
  const int lane  = threadIdx.x & 31;
  const int wave  = threadIdx.x >> 5;
  const int m0    = blockIdx.x << 4;
  const int n0    = wave << 4;
  const int l15   = lane & 15;
  const int khalf = (lane >> 4) << 1;
  const int mrow  = m0 + l15;
  const int ncol  = n0 + l15;

  v8f acc = {};

#pragma unroll
  for (int k0 = 0; k0 < DD; k0 += 4) {
    const int krow = k0 + khalf;
    v2f a;
    const float* ap = X + mrow * DD + krow;
    a.x = ap[0];
    a.y = ap[1];
    v2f b;
    b.x = W[krow * DD + ncol];
    b.y = W[(krow + 1) * DD + ncol];
    acc = wmma_f32_k4(a, b, acc);
  }

  const int mbase = m0 + ((lane >> 4) << 3);
#pragma unroll
  for (int i = 0; i < 8; ++i) {
    const float v = acc[i];
    Yout[(mbase + i) * DD + ncol] = 1.f / (1.f + __expf(-v));
  }
}

// ---------------------------------------------------------------------------
extern "C" void kernel_launch(void* const* d_in, const int* in_sizes, int n_in,
                              void* d_out, int out_size, void* d_ws, size_t ws_size,
                              hipStream_t stream) {
  const float* data = (const float*)d_in[0];   // (B,T,R,D) fp32
  const int*   nidx = (const int*)  d_in[1];   // (R,K) int32
  const float* wq   = (const float*)d_in[2];   // (D,D)
  const float* wk   = (const float*)d_in[3];   // (D,D)
  const float* wds  = (const float*)d_in[4];   // (D,D)
  float* out = (float*)d_out;                  // (B,T,R,D) fp32

  // Workspace: Q buffer (reused in-place for out=q+ctx) + Kp buffer.
  // Requires 2 * NROWS * D * 4 bytes = 50 MB of d_ws.
  float* qbuf = (float*)d_ws;
  float* kbuf = qbuf + (size_t)NROWS * DD;

  // 1) Q = X@Wq, Kp = X@Wk  — 6144 M-tiles, 4 waves/block (one per N tile)
  proj_qk_kernel<<<NROWS / 16, 128, 0, stream>>>(data, wq, wk, qbuf, kbuf);

  // 2) gather + scores + softmax + ctx, one wave per row, 8 waves/block
  attn_kernel<<<NROWS / 8, 256, 0, stream>>>(qbuf, kbuf, nidx);

  // 3) sigmoid((q+ctx) @ Wd) -> d_out
  out_proj_sigmoid_kernel<<<NROWS / 16, 128, 0, stream>>>(qbuf, wds, out);
}